// Permutation_8349416424143
// MI455X (gfx1250) — compile-verified
//
#include <hip/hip_runtime.h>
#include <math.h>

#define H 64
#define STRIDE 68   // padded LDS row stride (floats): 272B = 16B-aligned, conflict-friendly

typedef float v4f __attribute__((ext_vector_type(4)));

__global__ __launch_bounds__(256) void sinkhorn_node_kernel(const float* __restrict__ M,
                                                            float* __restrict__ out)
{
    __shared__ __align__(16) float Cs[H * STRIDE];   // 17408 B : the 64x64 block (padded)
    __shared__ float part[4 * H];                    //  1024 B : 4 row-chunk partials per column
    __shared__ __align__(16) float invcol[H];        //   256 B : 1/colsum broadcast

    const int t  = threadIdx.x;       // 0..255
    const int r  = t >> 2;            // row 0..63
    const int q  = t & 3;             // quarter index 0..3
    const int c0 = q << 4;            // first column of this thread's quarter-row
    const long node = blockIdx.x;

    const float* gsrc = M + node * (long)(H * H) + r * H + c0;
    const int lbase = r * STRIDE + c0;

    // ---- CDNA5 async copy: global -> LDS (ASYNCcnt), no VGPR staging ----
#pragma unroll
    for (int j4 = 0; j4 < 4; ++j4) {
        unsigned lds_off = (unsigned)(uintptr_t)(&Cs[lbase + 4 * j4]);
        unsigned long long ga = (unsigned long long)(gsrc + 4 * j4);
        asm volatile("global_load_async_to_lds_b128 %0, %1, off"
                     :: "v"(lds_off), "v"(ga) : "memory");
    }
    asm volatile("s_wait_asynccnt 0" ::: "memory");

    // Read back own raw values, apply C = exp(sigmoid(clamp(x, -10, 10)) / 0.01)
    // sigmoid via hardware v_exp_f32 + v_rcp_f32 (~2e-7 err in s); outer expf
    // kept fully precise as the accuracy anchor.
    float v[16];
#pragma unroll
    for (int j4 = 0; j4 < 4; ++j4) {
        v4f f = *(const v4f*)&Cs[lbase + 4 * j4];
#pragma unroll
        for (int k = 0; k < 4; ++k) {
            float x = fminf(fmaxf(f[k], -10.0f), 10.0f);
            float e = __builtin_amdgcn_exp2f(-x * 1.44269504088896f);  // e^{-x}
            float s = __builtin_amdgcn_rcpf(1.0f + e);                  // sigmoid
            v[4 * j4 + k] = expf(s * 100.0f);                           // 1/GAMMA = 100
        }
    }

    const int cc = t & 63;   // column this thread reduces
    const int rb = t >> 6;   // which 16-row chunk it reduces

    for (int it = 0; it < 5; ++it) {
        // publish current matrix to LDS
#pragma unroll
        for (int j4 = 0; j4 < 4; ++j4) {
            v4f f;
            f.x = v[4 * j4 + 0]; f.y = v[4 * j4 + 1];
            f.z = v[4 * j4 + 2]; f.w = v[4 * j4 + 3];
            *(v4f*)&Cs[lbase + 4 * j4] = f;
        }
        __syncthreads();

        // column partial sums: sum 16 rows of column cc (bank-conflict-free)
        float ps = 0.0f;
#pragma unroll
        for (int i = 0; i < 16; ++i)
            ps += Cs[(rb * 16 + i) * STRIDE + cc];
        part[rb * H + cc] = ps;
        __syncthreads();

        // combine 4 partials -> reciprocal of column sum (1-ulp hardware rcp)
        if (t < H) {
            float s = part[t] + part[H + t] + part[2 * H + t] + part[3 * H + t];
            invcol[t] = __builtin_amdgcn_rcpf(s);
        }
        __syncthreads();

        // column-normalize (registers) while accumulating the new row sum
        float rs = 0.0f;
#pragma unroll
        for (int j4 = 0; j4 < 4; ++j4) {
            v4f ic = *(const v4f*)&invcol[c0 + 4 * j4];
            v[4 * j4 + 0] *= ic.x; v[4 * j4 + 1] *= ic.y;
            v[4 * j4 + 2] *= ic.z; v[4 * j4 + 3] *= ic.w;
            rs += v[4 * j4 + 0] + v[4 * j4 + 1] + v[4 * j4 + 2] + v[4 * j4 + 3];
        }
        // 4 lanes (q = 0..3) share a row: reduce within the quad (wave32-safe)
        rs += __shfl_xor(rs, 1, 32);
        rs += __shfl_xor(rs, 2, 32);
        float rinv = __builtin_amdgcn_rcpf(rs);
#pragma unroll
        for (int j = 0; j < 16; ++j) v[j] *= rinv;
    }

    // write result once: coalesced, non-temporal b128 stores (write-once data,
    // keep it out of L2 so the streaming input keeps the cache)
    v4f* gdst = (v4f*)(out + node * (long)(H * H) + r * H + c0);
#pragma unroll
    for (int j4 = 0; j4 < 4; ++j4) {
        v4f f;
        f.x = v[4 * j4 + 0]; f.y = v[4 * j4 + 1];
        f.z = v[4 * j4 + 2]; f.w = v[4 * j4 + 3];
        __builtin_nontemporal_store(f, gdst + j4);
    }
}

extern "C" void kernel_launch(void* const* d_in, const int* in_sizes, int n_in,
                              void* d_out, int out_size, void* d_ws, size_t ws_size,
                              hipStream_t stream)
{
    (void)n_in; (void)d_ws; (void)ws_size; (void)out_size;
    const float* M = (const float*)d_in[0];
    float* out = (float*)d_out;
    const int n_nodes = in_sizes[0] / (H * H);   // 50000
    dim3 grid(n_nodes), block(256);
    sinkhorn_node_kernel<<<grid, block, 0, stream>>>(M, out);
}